// AffineFlow_44650480009355
// MI455X (gfx1250) — compile-verified
//
#include <hip/hip_runtime.h>

// CDNA5 / gfx1250: wave32, WMMA 16x16x4 f32.
typedef float v2f __attribute__((ext_vector_type(2)));
typedef float v8f __attribute__((ext_vector_type(8)));

namespace {

__device__ __forceinline__ void mm4(const float* a, const float* b, float* c) {
#pragma unroll
  for (int i = 0; i < 4; ++i) {
#pragma unroll
    for (int j = 0; j < 4; ++j) {
      float acc = 0.0f;
#pragma unroll
      for (int k = 0; k < 4; ++k) acc = fmaf(a[i * 4 + k], b[k * 4 + j], acc);
      c[i * 4 + j] = acc;
    }
  }
}

}  // namespace

// ---------------------------------------------------------------------------
// Phase 1: E_t = expm(t * [[M+M0, b],[0,0]]), one 4x4 per thread
// (scaling-and-squaring + order-12 Taylor/Horner). The result is written to
// ws ALREADY in the phase-2 WMMA A-operand lane layout:
//   Etab[tc*64 + lane*2 + {0,1}]  with lane = hi*16 + (tl*4 + ir),
//   holding E[t = tc*4 + tl][row ir][K = 2*hi + {0,1}]
// so each phase-2 per-chunk wave load is one contiguous 256B burst.
// ---------------------------------------------------------------------------
__global__ void affine_expm_kernel(const float* __restrict__ t,
                                   const float* __restrict__ M,
                                   const float* __restrict__ M0,
                                   const float* __restrict__ b,
                                   float* __restrict__ Etab, int T) {
  const int tid = blockIdx.x * blockDim.x + threadIdx.x;
  if (tid >= T) return;
  const float tv = t[tid];

  float A[16];
#pragma unroll
  for (int i = 0; i < 16; ++i) A[i] = 0.0f;
#pragma unroll
  for (int i = 0; i < 3; ++i) {
#pragma unroll
    for (int j = 0; j < 3; ++j) A[i * 4 + j] = (M[i * 3 + j] + M0[i * 3 + j]) * tv;
    A[i * 4 + 3] = b[i] * tv;
  }

  // inf-norm -> squarings so the Taylor argument has norm <= 0.25
  float nrm = 0.0f;
#pragma unroll
  for (int i = 0; i < 4; ++i) {
    const float r = fabsf(A[i * 4 + 0]) + fabsf(A[i * 4 + 1]) +
                    fabsf(A[i * 4 + 2]) + fabsf(A[i * 4 + 3]);
    nrm = fmaxf(nrm, r);
  }
  int s = 0;
  while (nrm > 0.25f && s < 40) { nrm *= 0.5f; ++s; }
  const float sc = ldexpf(1.0f, -s);
#pragma unroll
  for (int i = 0; i < 16; ++i) A[i] *= sc;

  // P = I + A(I + A/2 (I + A/3 (... (I + A/12))))
  float P[16], Tm[16];
#pragma unroll
  for (int i = 0; i < 16; ++i) P[i] = (i % 5 == 0) ? 1.0f : 0.0f;
  for (int k = 12; k >= 1; --k) {
    mm4(A, P, Tm);
    const float inv = 1.0f / (float)k;
#pragma unroll
    for (int i = 0; i < 16; ++i) P[i] = Tm[i] * inv + ((i % 5 == 0) ? 1.0f : 0.0f);
  }
  for (int q = 0; q < s; ++q) {
    mm4(P, P, Tm);
#pragma unroll
    for (int i = 0; i < 16; ++i) P[i] = Tm[i];
  }

  // Scatter into the WMMA A-operand layout (8 aligned float2 stores).
  const int tc = tid >> 2;  // chunk of 4 t-values
  const int tl = tid & 3;   // t within chunk
#pragma unroll
  for (int hi = 0; hi < 2; ++hi) {
#pragma unroll
    for (int ir = 0; ir < 4; ++ir) {
      float2 v;
      v.x = P[ir * 4 + 2 * hi + 0];
      v.y = P[ir * 4 + 2 * hi + 1];
      *reinterpret_cast<float2*>(Etab + tc * 64 + (hi * 16 + tl * 4 + ir) * 2) = v;
    }
  }
}

// ---------------------------------------------------------------------------
// Phase 2 (store-bandwidth bound, ~384 MB): per wave, one 16-point tile.
// One V_WMMA_F32_16X16X4_F32 per 4 t-values:
//   A[m,k] = E[t_base + (m>>2)][m&3][k]   (16x4, m packs 4 t x 4 rows)
//   B[k,n] = x_pad[n_base+n][k]           (4x16, k=3 coordinate is 0)
//   D[m,n] = out[t, n, i], i = m&3 (rows i==3 are identically 0 -> skipped)
// D layout: lane = column (point), VGPR v holds row m = v + 8*(lane>=16):
// each lane stores two contiguous 12B chunks -> dense 192B regions per t.
// ---------------------------------------------------------------------------
template <bool GUARD>
__device__ __forceinline__ void run_chunks16(const float* __restrict__ Etab,
                                             float* __restrict__ out,
                                             v2f Bv, int lane, int n, int N) {
  const int hi = lane >> 4;

  // Batch all 16 A-operand loads (one contiguous 256B burst per chunk).
  float2 e[16];
#pragma unroll
  for (int tc = 0; tc < 16; ++tc)
    e[tc] = *reinterpret_cast<const float2*>(Etab + tc * 64 + lane * 2);

  size_t baseA = ((size_t)(2 * hi) * (size_t)N + (size_t)n) * 3;
  const size_t strideT = (size_t)N * 3;
#pragma unroll
  for (int tc = 0; tc < 16; ++tc) {
    v2f Av;
    Av[0] = e[tc].x;
    Av[1] = e[tc].y;
    v8f c = {};
    // (neg_a, A, neg_b, B, c_mod, C, reuse_a, reuse_b)
    c = __builtin_amdgcn_wmma_f32_16x16x4_f32(false, Av, false, Bv,
                                              (short)0, c, false, false);
    if (!GUARD || n < N) {
      out[baseA + 0] = c[0];            // (t0, n, 0..2): v=0,1,2 -> b96
      out[baseA + 1] = c[1];
      out[baseA + 2] = c[2];
      const size_t baseB = baseA + strideT;
      out[baseB + 0] = c[4];            // (t0+1, n, 0..2): v=4,5,6 -> b96
      out[baseB + 1] = c[5];
      out[baseB + 2] = c[6];
      // v=3, v=7 are i==3 rows: identically zero, not part of the output.
    }
    baseA += strideT * 4;               // next chunk: t advances by 4
  }
}

// Specialized for T == 64 (16 fully unrolled chunks).
__global__ void affine_apply_wmma64(const float* __restrict__ x,
                                    const float* __restrict__ Etab,
                                    float* __restrict__ out, int N) {
  const int lane = threadIdx.x & 31;
  const int wave = threadIdx.x >> 5;
  const int ntiles = (N + 15) >> 4;
  const int tile = blockIdx.x * (blockDim.x >> 5) + wave;
  if (tile >= ntiles) return;  // wave-uniform: surviving waves keep EXEC all-1s

  const int lm = lane & 15;
  const int hi = lane >> 4;
  const int n = tile * 16 + lm;
  const int nc = (n < N) ? n : (N - 1);

  __builtin_prefetch(Etab + lane * 32, 0, 0);  // warm the 4KB expm table

  // B operand: lane holds B[K = 2*hi + vreg, N = lm] (mirrors A-layout rule).
  v2f Bv;
  Bv[0] = x[(size_t)nc * 3 + 2 * hi];                 // k = 0 or 2
  Bv[1] = hi ? 0.0f : x[(size_t)nc * 3 + 1];          // k = 1 or 3 (pad -> 0)

  if (tile * 16 + 16 <= N)
    run_chunks16<false>(Etab, out, Bv, lane, n, N);   // hot path: no exec play
  else
    run_chunks16<true>(Etab, out, Bv, lane, n, N);    // tail tile only
}

// Generic fallback for arbitrary T (runtime chunk loop, per-store t guards).
__global__ void affine_apply_wmma_gen(const float* __restrict__ x,
                                      const float* __restrict__ Etab,
                                      float* __restrict__ out, int N, int T) {
  const int lane = threadIdx.x & 31;
  const int wave = threadIdx.x >> 5;
  const int ntiles = (N + 15) >> 4;
  const int tile = blockIdx.x * (blockDim.x >> 5) + wave;
  if (tile >= ntiles) return;

  const int lm = lane & 15;
  const int hi = lane >> 4;
  const int n = tile * 16 + lm;
  const int nc = (n < N) ? n : (N - 1);

  v2f Bv;
  Bv[0] = x[(size_t)nc * 3 + 2 * hi];
  Bv[1] = hi ? 0.0f : x[(size_t)nc * 3 + 1];

  const int nchunks = (T + 3) >> 2;
  for (int tc = 0; tc < nchunks; ++tc) {
    const float2 e2 =
        *reinterpret_cast<const float2*>(Etab + tc * 64 + lane * 2);
    v2f Av;
    Av[0] = e2.x;
    Av[1] = e2.y;
    v8f c = {};
    c = __builtin_amdgcn_wmma_f32_16x16x4_f32(false, Av, false, Bv,
                                              (short)0, c, false, false);
    const int t0 = tc * 4 + 2 * hi;
    if (n < N && t0 < T) {
      const size_t baseA = ((size_t)t0 * (size_t)N + (size_t)n) * 3;
      out[baseA + 0] = c[0];
      out[baseA + 1] = c[1];
      out[baseA + 2] = c[2];
      if (t0 + 1 < T) {
        const size_t baseB = baseA + (size_t)N * 3;
        out[baseB + 0] = c[4];
        out[baseB + 1] = c[5];
        out[baseB + 2] = c[6];
      }
    }
  }
}

extern "C" void kernel_launch(void* const* d_in, const int* in_sizes, int n_in,
                              void* d_out, int out_size, void* d_ws, size_t ws_size,
                              hipStream_t stream) {
  const float* x  = (const float*)d_in[0];   // (N, 3)
  const float* t  = (const float*)d_in[1];   // (T,)
  const float* M  = (const float*)d_in[2];   // (3, 3)
  const float* M0 = (const float*)d_in[3];   // (3, 3)
  const float* b  = (const float*)d_in[4];   // (3,)
  float* out = (float*)d_out;                // (T, N, 3)
  float* Etab = (float*)d_ws;                // T * 16 floats (A-operand layout)

  const int N = in_sizes[0] / 3;
  const int T = in_sizes[1];

  // Phase 1: 64 tiny expm's.
  affine_expm_kernel<<<(T + 63) / 64, 64, 0, stream>>>(t, M, M0, b, Etab, T);

  // Phase 2: streaming WMMA apply. 8 waves/block, one 16-point tile per wave.
  const int ntiles = (N + 15) / 16;
  const int wavesPerBlock = 8;
  dim3 block(32 * wavesPerBlock);
  dim3 grid((ntiles + wavesPerBlock - 1) / wavesPerBlock);
  if (T == 64)
    affine_apply_wmma64<<<grid, block, 0, stream>>>(x, Etab, out, N);
  else
    affine_apply_wmma_gen<<<grid, block, 0, stream>>>(x, Etab, out, N, T);
}